// LightweightSSM_58488864637373
// MI455X (gfx1250) — compile-verified
//
#include <hip/hip_runtime.h>
#include <hip/hip_bf16.h>

// Sizes from the reference
#define D_MODEL 512
#define D_STATE 64
#define D_INNER 1024
#define DT_RANK 32
#define BATCH   2
#define SEQLEN  1024
#define M_TOK   (BATCH * SEQLEN)   // 2048 tokens
#define PROJ_W  (DT_RANK + 2 * D_STATE)  // 160

typedef __attribute__((ext_vector_type(2))) float    v2f;
typedef __attribute__((ext_vector_type(8))) float    v8f;
typedef __attribute__((ext_vector_type(4))) unsigned v4u;
typedef __attribute__((ext_vector_type(4))) int      v4i;
typedef __attribute__((ext_vector_type(8))) int      v8i;

__device__ __forceinline__ float silu_f(float x) {
    return x * __builtin_amdgcn_rcpf(1.f + __expf(-x));   // v_exp + v_rcp, no IEEE div
}
__device__ __forceinline__ float softplus_f(float x) {
    return (x > 20.f) ? x : __logf(1.f + __expf(x));
}

// ---------------------------------------------------------------------------
// TDM: issue a 2D tile load Global->LDS via the Tensor Data Mover.
//   tile = tile_rows x 32 f32 elements, row pitch in memory = ld elements.
//   LDS layout: row-major with pad after every 32 DWORDs of +2 DWORDs
//   (pad_interval=4 -> 1<<4 8-byte units = 128B; pad_amount=1 -> 2 DWORDs),
//   i.e. LDS pitch = 34 floats: 8B-aligned fragments, bank-conflict-free.
// Tracked by TENSORcnt; caller waits with s_wait_tensorcnt.
// This toolchain's builtin is the 6-arg form:
//   (v4u g0, v8i g1, v4i g2, v4i g3, v8i extra, i32 cpol)
// ---------------------------------------------------------------------------
#define LDS_PITCH 34
__device__ __forceinline__ void tdm_load_2d(unsigned lds_off, const float* gptr,
                                            unsigned tile_rows, unsigned ld,
                                            unsigned nrows) {
    const unsigned long long ga = (unsigned long long)(uintptr_t)gptr;
    v4u g0;
    g0.x = 1u;                                           // count=1 (valid user D#)
    g0.y = lds_off;                                      // lds_addr
    g0.z = (unsigned)ga;                                 // global_addr[31:0]
    g0.w = (unsigned)((ga >> 32) & 0x01FFFFFFu) | (2u << 30); // addr[56:32] | type=2
    v8i g1;
    g1[0] = (int)((2u << 16) | (1u << 20) | (4u << 22) | (1u << 25)); // 4B, pad on
    g1[1] = (int)((ld & 0xFFFFu) << 16);                 // tensor_dim0 lo16
    g1[2] = (int)((ld >> 16) | ((nrows & 0xFFFFu) << 16)); // dim0 hi | dim1 lo
    g1[3] = (int)((nrows >> 16) | (32u << 16));          // dim1 hi | tile_dim0=32
    g1[4] = (int)tile_rows;                              // tile_dim1 (tile_dim2=0)
    g1[5] = (int)ld;                                     // tensor_dim0_stride lo32
    g1[6] = 0;                                           // stride hi | dim1_stride lo
    g1[7] = 0;
    const v4i z4 = {0, 0, 0, 0};                         // groups 2/3 unused (2D)
    const v8i z8 = {0, 0, 0, 0, 0, 0, 0, 0};
    __builtin_amdgcn_tensor_load_to_lds(g0, g1, z4, z4, z8, 0);
}

// ---------------------------------------------------------------------------
// Register-blocked fp32 WMMA fed by direct global loads (for the small GEMMs).
// C[tm..,tn..] += A(MxK,row-major,ld=LDA) @ B(NxK,row-major,ld=LDB)^T
// ---------------------------------------------------------------------------
template <int K, int LDA, int LDB, int MT, int NT>
__device__ __forceinline__ void wmma_block(const float* __restrict__ A,
                                           const float* __restrict__ B,
                                           int tm, int tn, int lane,
                                           v8f c[MT][NT]) {
    const int r  = lane & 15;
    const int kk = (lane >> 4) << 1;          // 0 | 2
    const float* ap[MT];
    const float* bp[NT];
#pragma unroll
    for (int mi = 0; mi < MT; ++mi)
        ap[mi] = A + (size_t)(tm + mi * 16 + r) * LDA + kk;
#pragma unroll
    for (int ni = 0; ni < NT; ++ni)
        bp[ni] = B + (size_t)(tn + ni * 16 + r) * LDB + kk;
#pragma unroll
    for (int mi = 0; mi < MT; ++mi)
#pragma unroll
        for (int ni = 0; ni < NT; ++ni)
            c[mi][ni] = (v8f){0.f, 0.f, 0.f, 0.f, 0.f, 0.f, 0.f, 0.f};

#pragma unroll 1
    for (int k0 = 0; k0 < K; k0 += 16) {
        if (k0 + 16 < K) {
#pragma unroll
            for (int mi = 0; mi < MT; ++mi) __builtin_prefetch(ap[mi] + k0 + 16, 0, 3);
#pragma unroll
            for (int ni = 0; ni < NT; ++ni) __builtin_prefetch(bp[ni] + k0 + 16, 0, 3);
        }
#pragma unroll
        for (int k1 = 0; k1 < 16; k1 += 4) {
            v2f a[MT], b[NT];
#pragma unroll
            for (int mi = 0; mi < MT; ++mi) a[mi] = *(const v2f*)(ap[mi] + k0 + k1);
#pragma unroll
            for (int ni = 0; ni < NT; ++ni) b[ni] = *(const v2f*)(bp[ni] + k0 + k1);
#pragma unroll
            for (int mi = 0; mi < MT; ++mi)
#pragma unroll
                for (int ni = 0; ni < NT; ++ni)
                    c[mi][ni] = __builtin_amdgcn_wmma_f32_16x16x4_f32(
                        false, a[mi], false, b[ni], (short)0, c[mi][ni], false, false);
        }
    }
}

// ---------------------------------------------------------------------------
// Stage 1: LayerNorm over D_MODEL=512 (one block of 256 threads per token)
// ---------------------------------------------------------------------------
__global__ void ln_kernel(const float* __restrict__ x, const float* __restrict__ w,
                          const float* __restrict__ b, float* __restrict__ h) {
    __shared__ float red[256];
    const int row = blockIdx.x;
    const int t   = threadIdx.x;
    const float* xr = x + (size_t)row * D_MODEL;
    float v0 = xr[t], v1 = xr[t + 256];
    red[t] = v0 + v1;
    __syncthreads();
#pragma unroll
    for (int s = 128; s > 0; s >>= 1) {
        if (t < s) red[t] += red[t + s];
        __syncthreads();
    }
    const float mu = red[0] * (1.f / (float)D_MODEL);
    __syncthreads();
    const float d0 = v0 - mu, d1 = v1 - mu;
    red[t] = d0 * d0 + d1 * d1;
    __syncthreads();
#pragma unroll
    for (int s = 128; s > 0; s >>= 1) {
        if (t < s) red[t] += red[t + s];
        __syncthreads();
    }
    const float rstd = rsqrtf(red[0] * (1.f / (float)D_MODEL) + 1e-5f);
    float* hr = h + (size_t)row * D_MODEL;
    hr[t]       = d0 * rstd * w[t]       + b[t];
    hr[t + 256] = d1 * rstd * w[t + 256] + b[t + 256];
}

// ---------------------------------------------------------------------------
// Stage 2: xz = h @ W_in^T, fused silu + split   (M=2048, N=2048, K=512)
// TDM-staged, double-buffered LDS pipeline:
//   block tile = 128M x 64N, K-slab = 32.
//   Wave 0 issues both TENSOR_LOAD_TO_LDS and waits TENSORcnt; barriers
//   publish the slab; slab k+1 DMA overlaps compute of slab k.
//   8 waves = 4(M) x 2(N), each 2x2 WMMA tiles (4 independent accum chains).
// ---------------------------------------------------------------------------
#define XZ_AOFF (128 * LDS_PITCH)            // B buffer offset (floats)
#define XZ_BUF  (XZ_AOFF + 64 * LDS_PITCH)   // floats per stage = 6528
__global__ void gemm_xz_kernel(const float* __restrict__ H, const float* __restrict__ Win,
                               float* __restrict__ x_in, float* __restrict__ sz) {
    __shared__ float smem[2][XZ_BUF];        // 52224 B
    const int tid  = threadIdx.x;
    const int lane = tid & 31;
    const int wave = tid >> 5;
    const int bm = blockIdx.x >> 5;          // 2048/64 = 32 col-blocks
    const int bn = blockIdx.x & 31;
    const int row0 = bm << 7;                // *128
    const int col0 = bn << 6;                // *64

    // prologue: DMA slab 0
    if (wave == 0) {
        tdm_load_2d((unsigned)(uintptr_t)&smem[0][0],       H   + (size_t)row0 * D_MODEL, 128, D_MODEL, M_TOK);
        tdm_load_2d((unsigned)(uintptr_t)&smem[0][XZ_AOFF], Win + (size_t)col0 * D_MODEL,  64, D_MODEL, 2 * D_INNER);
    }

    const int m_l = (wave >> 1) << 5;        // 0,32,64,96
    const int n_l = (wave & 1) << 5;         // 0,32
    const int r   = lane & 15;
    const int kko = (lane >> 4) << 1;
    v8f c[2][2];
#pragma unroll
    for (int mi = 0; mi < 2; ++mi)
#pragma unroll
        for (int ni = 0; ni < 2; ++ni)
            c[mi][ni] = (v8f){0.f, 0.f, 0.f, 0.f, 0.f, 0.f, 0.f, 0.f};

#pragma unroll 1
    for (int ks = 0; ks < D_MODEL / 32; ++ks) {
        if (wave == 0) __builtin_amdgcn_s_wait_tensorcnt(0);
        __syncthreads();                     // slab ks visible to all waves
        const int cur = ks & 1;
        if (wave == 0 && ks + 1 < D_MODEL / 32) {
            const int kn = (ks + 1) << 5;
            tdm_load_2d((unsigned)(uintptr_t)&smem[cur ^ 1][0],       H   + (size_t)row0 * D_MODEL + kn, 128, D_MODEL, M_TOK);
            tdm_load_2d((unsigned)(uintptr_t)&smem[cur ^ 1][XZ_AOFF], Win + (size_t)col0 * D_MODEL + kn,  64, D_MODEL, 2 * D_INNER);
        }
        const float* As = &smem[cur][(m_l + r) * LDS_PITCH + kko];
        const float* Bs = &smem[cur][XZ_AOFF + (n_l + r) * LDS_PITCH + kko];
#pragma unroll
        for (int k1 = 0; k1 < 32; k1 += 4) {
            v2f a[2], b[2];
#pragma unroll
            for (int mi = 0; mi < 2; ++mi) a[mi] = *(const v2f*)(As + mi * 16 * LDS_PITCH + k1);
#pragma unroll
            for (int ni = 0; ni < 2; ++ni) b[ni] = *(const v2f*)(Bs + ni * 16 * LDS_PITCH + k1);
#pragma unroll
            for (int mi = 0; mi < 2; ++mi)
#pragma unroll
                for (int ni = 0; ni < 2; ++ni)
                    c[mi][ni] = __builtin_amdgcn_wmma_f32_16x16x4_f32(
                        false, a[mi], false, b[ni], (short)0, c[mi][ni], false, false);
        }
        __syncthreads();                     // slab fully consumed before reuse
    }

    // epilogue: silu + split store
#pragma unroll
    for (int mi = 0; mi < 2; ++mi) {
        const int rb = row0 + m_l + mi * 16 + ((lane >> 4) << 3);
#pragma unroll
        for (int ni = 0; ni < 2; ++ni) {
            const int col = col0 + n_l + ni * 16 + (lane & 15);
            float* dst = (col < D_INNER) ? (x_in + col) : (sz + (col - D_INNER));
#pragma unroll
            for (int rr = 0; rr < 8; ++rr)
                dst[(size_t)(rb + rr) * D_INNER] = silu_f(c[mi][ni][rr]);
        }
    }
}

// ---------------------------------------------------------------------------
// Stage 3: proj = x_in @ W_x^T   (M=2048, N=160, K=1024), 4x1 tiles/wave
// ---------------------------------------------------------------------------
__global__ void gemm_proj_kernel(const float* __restrict__ Xin, const float* __restrict__ Wx,
                                 float* __restrict__ proj) {
    const int lane = threadIdx.x & 31;
    const int wid  = blockIdx.x * 8 + (threadIdx.x >> 5);
    const int tm = (wid / 10) << 6;
    const int tn = (wid % 10) << 4;
    v8f c[4][1];
    wmma_block<D_INNER, D_INNER, D_INNER, 4, 1>(Xin, Wx, tm, tn, lane, c);
    const int col = tn + (lane & 15);
#pragma unroll
    for (int mi = 0; mi < 4; ++mi) {
        const int rb = tm + mi * 16 + ((lane >> 4) << 3);
#pragma unroll
        for (int r = 0; r < 8; ++r)
            proj[(size_t)(rb + r) * PROJ_W + col] = c[mi][0][r];
    }
}

// ---------------------------------------------------------------------------
// Stage 4: dt = softplus(proj[:, :32] @ W_dt^T + b_dt)  (M=2048, N=1024, K=32)
// ---------------------------------------------------------------------------
__global__ void gemm_dt_kernel(const float* __restrict__ proj, const float* __restrict__ Wdt,
                               const float* __restrict__ bdt, float* __restrict__ dt) {
    const int lane = threadIdx.x & 31;
    const int wid  = blockIdx.x * 8 + (threadIdx.x >> 5);
    const int tm = (wid >> 5) << 5;
    const int tn = (wid & 31) << 5;
    v8f c[2][2];
    wmma_block<DT_RANK, PROJ_W, DT_RANK, 2, 2>(proj, Wdt, tm, tn, lane, c);
#pragma unroll
    for (int mi = 0; mi < 2; ++mi) {
        const int rb = tm + mi * 16 + ((lane >> 4) << 3);
#pragma unroll
        for (int ni = 0; ni < 2; ++ni) {
            const int col = tn + ni * 16 + (lane & 15);
            const float bv = bdt[col];
#pragma unroll
            for (int r = 0; r < 8; ++r)
                dt[(size_t)(rb + r) * D_INNER + col] = softplus_f(c[mi][ni][r] + bv);
        }
    }
}

// ---------------------------------------------------------------------------
// Stage 5a: negA = -exp(A_log)
// ---------------------------------------------------------------------------
__global__ void nega_kernel(const float* __restrict__ A_log, float* __restrict__ negA) {
    const int i = blockIdx.x * 256 + threadIdx.x;
    negA[i] = -__expf(A_log[i]);
}

// ---------------------------------------------------------------------------
// Stage 5b: exp-dominated mix (TRANS-bound: 134M v_exp_f32)
// y[m,i] = (sum_d exp(dt[m,i]*negA[i,d]) * BC[m,d]) * x_in * silu(z) + x_in*D
// ---------------------------------------------------------------------------
__global__ void mix_kernel(const float* dt, const float* __restrict__ proj,
                           const float* __restrict__ negA, const float* __restrict__ x_in,
                           const float* __restrict__ sz, const float* __restrict__ Dv,
                           float* y) {
    __shared__ float bc[D_STATE];
    const int m = blockIdx.x;
    const int t = threadIdx.x;
    if (t < D_STATE) {
        const float* pr = proj + (size_t)m * PROJ_W;
        bc[t] = pr[DT_RANK + t] * pr[DT_RANK + D_STATE + t];
    }
    __syncthreads();
#pragma unroll
    for (int ii = 0; ii < 4; ++ii) {
        const int i = t + ii * 256;
        const size_t off = (size_t)m * D_INNER + i;
        const float dtv = dt[off];
        const float* Ar = negA + (size_t)i * D_STATE;
        float acc = 0.f;
#pragma unroll 8
        for (int d = 0; d < D_STATE; ++d)
            acc += __expf(dtv * Ar[d]) * bc[d];
        const float xi = x_in[off];
        y[off] = acc * xi * sz[off] + xi * Dv[i];
    }
}

// ---------------------------------------------------------------------------
// Stage 6: out = skip + y @ W_out^T  (M=2048, N=512, K=1024), 2x2 tiles/wave
// ---------------------------------------------------------------------------
__global__ void gemm_out_kernel(const float* __restrict__ Y, const float* __restrict__ Wout,
                                const float* __restrict__ skip, float* __restrict__ out) {
    const int lane = threadIdx.x & 31;
    const int wid  = blockIdx.x * 8 + (threadIdx.x >> 5);
    const int tm = (wid >> 4) << 5;
    const int tn = (wid & 15) << 5;
    v8f c[2][2];
    wmma_block<D_INNER, D_INNER, D_INNER, 2, 2>(Y, Wout, tm, tn, lane, c);
#pragma unroll
    for (int mi = 0; mi < 2; ++mi) {
        const int rb = tm + mi * 16 + ((lane >> 4) << 3);
#pragma unroll
        for (int ni = 0; ni < 2; ++ni) {
            const int col = tn + ni * 16 + (lane & 15);
#pragma unroll
            for (int r = 0; r < 8; ++r) {
                const size_t o = (size_t)(rb + r) * D_MODEL + col;
                out[o] = skip[o] + c[mi][ni][r];
            }
        }
    }
}

// ---------------------------------------------------------------------------
extern "C" void kernel_launch(void* const* d_in, const int* in_sizes, int n_in,
                              void* d_out, int out_size, void* d_ws, size_t ws_size,
                              hipStream_t stream) {
    const float* x      = (const float*)d_in[0];
    const float* norm_w = (const float*)d_in[1];
    const float* norm_b = (const float*)d_in[2];
    const float* W_in   = (const float*)d_in[3];
    const float* W_x    = (const float*)d_in[4];
    const float* W_dt   = (const float*)d_in[5];
    const float* b_dt   = (const float*)d_in[6];
    const float* A_log  = (const float*)d_in[7];
    const float* Dv     = (const float*)d_in[8];
    const float* W_out  = (const float*)d_in[9];
    float* out = (float*)d_out;

    float* ws   = (float*)d_ws;
    float* h    = ws;                               // 2048*512
    float* x_in = h    + (size_t)M_TOK * D_MODEL;   // 2048*1024
    float* sz   = x_in + (size_t)M_TOK * D_INNER;   // 2048*1024
    float* proj = sz   + (size_t)M_TOK * D_INNER;   // 2048*160
    float* dt   = proj + (size_t)M_TOK * PROJ_W;    // 2048*1024 (reused as y)
    float* negA = dt   + (size_t)M_TOK * D_INNER;   // 1024*64

    ln_kernel<<<M_TOK, 256, 0, stream>>>(x, norm_w, norm_b, h);
    nega_kernel<<<(D_INNER * D_STATE) / 256, 256, 0, stream>>>(A_log, negA);
    gemm_xz_kernel<<<512, 256, 0, stream>>>(h, W_in, x_in, sz);   // 16 x 32 block tiles
    gemm_proj_kernel<<<40, 256, 0, stream>>>(x_in, W_x, proj);
    gemm_dt_kernel<<<256, 256, 0, stream>>>(proj, W_dt, b_dt, dt);
    mix_kernel<<<M_TOK, 256, 0, stream>>>(dt, proj, negA, x_in, sz, Dv, dt);
    gemm_out_kernel<<<128, 256, 0, stream>>>(dt, W_out, x, out);
}